// HeatmapQueryGenerator_62397284876860
// MI455X (gfx1250) — compile-verified
//
#include <hip/hip_runtime.h>
#include <math.h>

typedef __attribute__((ext_vector_type(2))) float v2f;
typedef __attribute__((ext_vector_type(8))) float v8f;

#define HIDN 256
#define NUMQ 100
#define HMH 136
#define HMW 136
#define NB 8
#define NHM (HMH*HMW)          /* 18496 */
#define THRESH 0.1f

/* workspace layout (float offsets) */
#define WS_COORDS 0                         /* 800*2  */
#define WS_POOLED 1600                      /* 800*256 */
#define WS_MASKED (WS_POOLED + 800*HIDN)    /* 8*18496 */
#define WS_FLAT   (WS_MASKED + NB*NHM)      /* 8*18496 */
#define WS_MV     (WS_FLAT + NB*NHM)        /* 800 */
#define WS_MI     (WS_MV + 800)             /* 800 (int) */
#define WS_FV     (WS_MI + 800)             /* 800 */
#define WS_FI     (WS_FV + 800)             /* 800 (int) */
#define WS_CNT    (WS_FI + 800)             /* 8 (int) */

/* output layout (float offsets): qe, content, qpos, coords, scores */
#define OUT_QE      0
#define OUT_CONTENT 409600
#define OUT_QPOS    614400
#define OUT_COORDS  819200
#define OUT_SCORES  820800

__device__ __forceinline__ v8f vzero8() {
  v8f z = {0.f,0.f,0.f,0.f,0.f,0.f,0.f,0.f};
  return z;
}

__device__ __forceinline__ v8f wmma_f32(v2f a, v2f b, v8f c) {
  /* D = A(16x4,f32) * B(4x16,f32) + C(16x16,f32) */
  return __builtin_amdgcn_wmma_f32_16x16x4_f32(false, a, false, b, (short)0, c, false, false);
}

/* ---------------- stage 0: init counts ---------------- */
__global__ void k_init(int* cnt) {
  if (threadIdx.x < NB) cnt[threadIdx.x] = 0;
}

/* ---------------- stage 1: NMS mask + copies ---------------- */
__global__ void k_nms(const float* __restrict__ hm, float* __restrict__ masked,
                      float* __restrict__ flat, int* __restrict__ cnt) {
  int gid = blockIdx.x * blockDim.x + threadIdx.x;
  if (gid >= NB * NHM) return;
  int b = gid / NHM, p = gid % NHM;
  int y = p / HMW, x = p % HMW;
  const float* h = hm + (size_t)b * NHM;
  float v = h[p];
  float mx = 0.0f;  /* reference zero-pads before max */
  for (int dy = -2; dy <= 2; ++dy)
    for (int dx = -2; dx <= 2; ++dx) {
      int yy = y + dy, xx = x + dx;
      if (yy >= 0 && yy < HMH && xx >= 0 && xx < HMW)
        mx = fmaxf(mx, h[yy * HMW + xx]);
    }
  bool keep = (v == mx) && (v > THRESH);
  flat[gid]   = v;
  masked[gid] = keep ? v : -INFINITY;
  if (keep) atomicAdd(&cnt[b], 1);
}

/* ---------------- stage 2: iterative top-100 (tie -> lowest index) ------- */
__global__ void k_select(float* __restrict__ ws) {
  int b = blockIdx.x;
  int which = blockIdx.y;  /* 0 = masked, 1 = flat */
  float* arr  = ws + (which == 0 ? WS_MASKED : WS_FLAT) + (size_t)b * NHM;
  float* outv = ws + (which == 0 ? WS_MV : WS_FV) + b * NUMQ;
  int*   outi = (int*)(ws + (which == 0 ? WS_MI : WS_FI)) + b * NUMQ;
  __shared__ float sv[256];
  __shared__ int   si[256];
  int tid = threadIdx.x;
  for (int k = 0; k < NUMQ; ++k) {
    float bv = -INFINITY; int bi = NHM;
    for (int i = tid; i < NHM; i += 256) {
      float v = arr[i];
      if (v > bv || (v == bv && i < bi)) { bv = v; bi = i; }
    }
    sv[tid] = bv; si[tid] = bi;
    __syncthreads();
    for (int s = 128; s > 0; s >>= 1) {
      if (tid < s) {
        float ov = sv[tid + s]; int oi = si[tid + s];
        if (ov > sv[tid] || (ov == sv[tid] && oi < si[tid])) { sv[tid] = ov; si[tid] = oi; }
      }
      __syncthreads();
    }
    if (tid == 0) {
      outv[k] = sv[0]; outi[k] = si[0];
      if (si[0] < NHM) arr[si[0]] = -INFINITY;
    }
    __syncthreads();
  }
}

/* ---------------- stage 3: combine branches, coords + scores ------------- */
__global__ void k_combine(float* __restrict__ ws, float* __restrict__ out) {
  int b = blockIdx.x, k = threadIdx.x;
  if (k >= NUMQ) return;
  const int* cnt = (const int*)(ws + WS_CNT);
  bool cond = cnt[b] >= NUMQ;
  float x, y, sc;
  if (cond) {
    float v = ws[WS_MV + b * NUMQ + k];
    int idx  = ((const int*)(ws + WS_MI))[b * NUMQ + k];
    bool fin = (v != -INFINITY);
    sc = fin ? v : 0.0f;
    x = fin ? (float)(idx % HMW) : 0.0f;
    y = fin ? (float)(idx / HMW) : 0.0f;
  } else {
    float v = ws[WS_FV + b * NUMQ + k];
    int idx  = ((const int*)(ws + WS_FI))[b * NUMQ + k];
    sc = v;
    x = (float)(idx % HMW);
    y = (float)(idx / HMW);
  }
  x = fminf(fmaxf(x / (float)(HMW - 1), 0.0f), 1.0f);
  y = fminf(fmaxf(y / (float)(HMH - 1), 0.0f), 1.0f);
  int r = b * NUMQ + k;
  ws[WS_COORDS + r * 2 + 0] = x;
  ws[WS_COORDS + r * 2 + 1] = y;
  out[OUT_COORDS + r * 2 + 0] = x;
  out[OUT_COORDS + r * 2 + 1] = y;
  out[OUT_SCORES + r] = sc;
}

/* ---------------- stage 4: multi-level weighted 3x3 pooling -------------- */
__global__ void __launch_bounds__(256)
k_pool(const float* __restrict__ s0, const float* __restrict__ s1,
       const float* __restrict__ s2, const float* __restrict__ s3,
       const float* __restrict__ hm, const float* __restrict__ lvlw_in,
       float* __restrict__ ws) {
  int k = blockIdx.x, b = blockIdx.y;
  int c = threadIdx.x;                 /* one thread per channel */
  int r = b * NUMQ + k;
  float cxn = ws[WS_COORDS + r * 2 + 0];
  float cyn = ws[WS_COORDS + r * 2 + 1];

  /* softmax over level weights */
  float w0 = lvlw_in[0], w1 = lvlw_in[1], w2 = lvlw_in[2], w3 = lvlw_in[3];
  float mw = fmaxf(fmaxf(w0, w1), fmaxf(w2, w3));
  float e0 = expf(w0 - mw), e1 = expf(w1 - mw), e2 = expf(w2 - mw), e3 = expf(w3 - mw);
  float es = e0 + e1 + e2 + e3;
  float lw[4] = { e0 / es, e1 / es, e2 / es, e3 / es };

  const float* srcs[4] = { s0, s1, s2, s3 };
  const int HL[4] = { 136, 68, 34, 17 };

  __shared__ float logit[9];
  __shared__ int cy9[9], cx9[9];

  const float* hb = hm + (size_t)b * NHM;
  float acc = 0.0f;

  for (int l = 0; l < 4; ++l) {
    int Hl = HL[l], Wl = HL[l];
    int cx = (int)(cxn * (float)(Wl - 1));
    int cy = (int)(cyn * (float)(Hl - 1));
    if (threadIdx.x < 9) {
      int i = threadIdx.x / 3, j = threadIdx.x % 3;
      int ysv = cy + i - 1, xsv = cx + j - 1;
      bool valid = (ysv >= 0) && (ysv < Hl) && (xsv >= 0) && (xsv < Wl);
      int ysc = min(max(ysv, 0), Hl - 1);
      int xsc = min(max(xsv, 0), Wl - 1);
      /* bilinear sample of full-res heatmap at align_corners level coords */
      float fy = (float)ysc * (float)(HMH - 1) / (float)(Hl - 1);
      float fx = (float)xsc * (float)(HMW - 1) / (float)(Wl - 1);
      int y0 = (int)floorf(fy); int y1 = min(y0 + 1, HMH - 1); float wy = fy - (float)y0;
      int x0 = (int)floorf(fx); int x1 = min(x0 + 1, HMW - 1); float wx = fx - (float)x0;
      float v00 = hb[y0 * HMW + x0], v01 = hb[y0 * HMW + x1];
      float v10 = hb[y1 * HMW + x0], v11 = hb[y1 * HMW + x1];
      float heat = (v00 * (1.f - wx) + v01 * wx) * (1.f - wy)
                 + (v10 * (1.f - wx) + v11 * wx) * wy;
      logit[threadIdx.x] = valid ? heat : -INFINITY;
      cy9[threadIdx.x] = ysc; cx9[threadIdx.x] = xsc;
    }
    __syncthreads();
    float mx = -INFINITY;
    for (int t = 0; t < 9; ++t) mx = fmaxf(mx, logit[t]);
    float ex[9]; float se = 0.0f;
    for (int t = 0; t < 9; ++t) { ex[t] = expf(logit[t] - mx); se += ex[t]; }
    const float* f = srcs[l] + ((size_t)b * HIDN + c) * (size_t)(Hl * Wl);
    float al = 0.0f;
    for (int t = 0; t < 9; ++t)
      al += (ex[t] / se) * f[cy9[t] * Wl + cx9[t]];
    acc += lw[l] * al;
    __syncthreads();
  }
  ws[WS_POOLED + (size_t)r * HIDN + c] = acc;
}

/* ---------------- stage 5: content MLP (WMMA f32 16x16x4) ---------------- */
__global__ void __launch_bounds__(256)
k_content(const float* __restrict__ ws, float* __restrict__ out,
          const float* __restrict__ wM1, const float* __restrict__ bV1,
          const float* __restrict__ gV1, const float* __restrict__ beV1,
          const float* __restrict__ wM2, const float* __restrict__ bV2) {
  __shared__ float As[16][HIDN];
  __shared__ float Hs[16][HIDN];
  __shared__ float rsum[16], rsq[16];
  int tid = threadIdx.x;
  int r0 = blockIdx.x * 16;

  for (int idx = tid; idx < 16 * HIDN; idx += 256) {
    int m = idx >> 8, n = idx & 255;
    As[m][n] = ws[WS_POOLED + (size_t)(r0 + m) * HIDN + n];
  }
  if (tid < 16) { rsum[tid] = 0.f; rsq[tid] = 0.f; }
  __syncthreads();

  int wave = tid >> 5, lane = tid & 31;
  int mrow = lane & 15;
  int kb = (lane >> 4) * 2;
  int na = wave * 32 + (lane & 15);
  int nb = wave * 32 + 16 + (lane & 15);
  int mbase = (lane >> 4) * 8;

  __builtin_prefetch(&wM1[(size_t)na * HIDN], 0, 1);
  __builtin_prefetch(&wM1[(size_t)nb * HIDN], 0, 1);

  v8f c0 = vzero8(), c1 = vzero8();
  for (int kk = 0; kk < HIDN; kk += 4) {
    v2f a;  a.x  = As[mrow][kk + kb];           a.y  = As[mrow][kk + kb + 1];
    v2f b0; b0.x = wM1[(size_t)na * HIDN + kk + kb]; b0.y = wM1[(size_t)na * HIDN + kk + kb + 1];
    v2f b1; b1.x = wM1[(size_t)nb * HIDN + kk + kb]; b1.y = wM1[(size_t)nb * HIDN + kk + kb + 1];
    c0 = wmma_f32(a, b0, c0);
    c1 = wmma_f32(a, b1, c1);
  }

  float va[8], vb[8];
  {
    float biasa = bV1[na], biasb = bV1[nb];
    for (int v = 0; v < 8; ++v) {
      va[v] = c0[v] + biasa;
      vb[v] = c1[v] + biasb;
      int m = mbase + v;
      atomicAdd(&rsum[m], va[v] + vb[v]);
      atomicAdd(&rsq[m],  va[v] * va[v] + vb[v] * vb[v]);
    }
  }
  __syncthreads();
  {
    float ga = gV1[na], bea = beV1[na], gb = gV1[nb], beb = beV1[nb];
    for (int v = 0; v < 8; ++v) {
      int m = mbase + v;
      float mean = rsum[m] * (1.0f / HIDN);
      float var  = rsq[m] * (1.0f / HIDN) - mean * mean;
      float inv  = rsqrtf(var + 1e-5f);
      Hs[m][na] = fmaxf((va[v] - mean) * inv * ga + bea, 0.0f);
      Hs[m][nb] = fmaxf((vb[v] - mean) * inv * gb + beb, 0.0f);
    }
  }
  __syncthreads();

  c0 = vzero8(); c1 = vzero8();
  for (int kk = 0; kk < HIDN; kk += 4) {
    v2f a;  a.x  = Hs[mrow][kk + kb];           a.y  = Hs[mrow][kk + kb + 1];
    v2f b0; b0.x = wM2[(size_t)na * HIDN + kk + kb]; b0.y = wM2[(size_t)na * HIDN + kk + kb + 1];
    v2f b1; b1.x = wM2[(size_t)nb * HIDN + kk + kb]; b1.y = wM2[(size_t)nb * HIDN + kk + kb + 1];
    c0 = wmma_f32(a, b0, c0);
    c1 = wmma_f32(a, b1, c1);
  }
  {
    float biasa = bV2[na], biasb = bV2[nb];
    for (int v = 0; v < 8; ++v) {
      int m = mbase + v;
      int r = r0 + m;
      float ca = c0[v] + biasa, cb = c1[v] + biasb;
      out[OUT_CONTENT + (size_t)r * HIDN + na] = ca;
      out[OUT_CONTENT + (size_t)r * HIDN + nb] = cb;
      out[OUT_QE + (size_t)r * 512 + 256 + na] = ca;
      out[OUT_QE + (size_t)r * 512 + 256 + nb] = cb;
    }
  }
}

/* ---------------- stage 6: positional path (WMMA f32 16x16x4) ------------ */
__global__ void __launch_bounds__(256)
k_pos(const float* __restrict__ ws, float* __restrict__ out,
      const float* __restrict__ pW, const float* __restrict__ pB,
      const float* __restrict__ pG, const float* __restrict__ pBE) {
  __shared__ float Ps[16][HIDN];
  __shared__ float rsum[16], rsq[16];
  int tid = threadIdx.x;
  int r0 = blockIdx.x * 16;
  const float TWO_PI = 6.283185307179586f;

  for (int idx = tid; idx < 16 * HIDN; idx += 256) {
    int m = idx >> 8, n = idx & 255;
    int r = r0 + m;
    float cx = ws[WS_COORDS + r * 2 + 0] * TWO_PI;
    float cy = ws[WS_COORDS + r * 2 + 1] * TWO_PI;
    int half = n & 127;
    float cc = (n < 128) ? cx : cy;
    int j = half >> 1;
    float t = __powf(10000.0f, (float)(2 * j) / 128.0f);
    float v = cc / t;
    Ps[m][n] = (half & 1) ? cosf(v) : sinf(v);
  }
  if (tid < 16) { rsum[tid] = 0.f; rsq[tid] = 0.f; }
  __syncthreads();

  int wave = tid >> 5, lane = tid & 31;
  int mrow = lane & 15;
  int kb = (lane >> 4) * 2;
  int na = wave * 32 + (lane & 15);
  int nb = wave * 32 + 16 + (lane & 15);
  int mbase = (lane >> 4) * 8;

  __builtin_prefetch(&pW[(size_t)na * HIDN], 0, 1);

  v8f c0 = vzero8(), c1 = vzero8();
  for (int kk = 0; kk < HIDN; kk += 4) {
    v2f a;  a.x  = Ps[mrow][kk + kb];          a.y  = Ps[mrow][kk + kb + 1];
    v2f b0; b0.x = pW[(size_t)na * HIDN + kk + kb]; b0.y = pW[(size_t)na * HIDN + kk + kb + 1];
    v2f b1; b1.x = pW[(size_t)nb * HIDN + kk + kb]; b1.y = pW[(size_t)nb * HIDN + kk + kb + 1];
    c0 = wmma_f32(a, b0, c0);
    c1 = wmma_f32(a, b1, c1);
  }

  float va[8], vb[8];
  {
    float biasa = pB[na], biasb = pB[nb];
    for (int v = 0; v < 8; ++v) {
      va[v] = c0[v] + biasa;
      vb[v] = c1[v] + biasb;
      int m = mbase + v;
      atomicAdd(&rsum[m], va[v] + vb[v]);
      atomicAdd(&rsq[m],  va[v] * va[v] + vb[v] * vb[v]);
    }
  }
  __syncthreads();
  {
    float ga = pG[na], bea = pBE[na], gb = pG[nb], beb = pBE[nb];
    for (int v = 0; v < 8; ++v) {
      int m = mbase + v;
      float mean = rsum[m] * (1.0f / HIDN);
      float var  = rsq[m] * (1.0f / HIDN) - mean * mean;
      float inv  = rsqrtf(var + 1e-5f);
      int r = r0 + m;
      float qa = (va[v] - mean) * inv * ga + bea;
      float qb = (vb[v] - mean) * inv * gb + beb;
      out[OUT_QPOS + (size_t)r * HIDN + na] = qa;
      out[OUT_QPOS + (size_t)r * HIDN + nb] = qb;
      out[OUT_QE + (size_t)r * 512 + na] = qa;
      out[OUT_QE + (size_t)r * 512 + nb] = qb;
    }
  }
}

/* ---------------- launcher ---------------- */
extern "C" void kernel_launch(void* const* d_in, const int* in_sizes, int n_in,
                              void* d_out, int out_size, void* d_ws, size_t ws_size,
                              hipStream_t stream) {
  const float* s0    = (const float*)d_in[0];
  const float* s1    = (const float*)d_in[1];
  const float* s2    = (const float*)d_in[2];
  const float* s3    = (const float*)d_in[3];
  /* d_in[4] spatial_shapes: known statically, unused */
  const float* hm    = (const float*)d_in[5];
  const float* lvlw  = (const float*)d_in[6];
  const float* c_w1  = (const float*)d_in[7];
  const float* c_b1  = (const float*)d_in[8];
  const float* c_g1  = (const float*)d_in[9];
  const float* c_be1 = (const float*)d_in[10];
  const float* c_w2  = (const float*)d_in[11];
  const float* c_b2  = (const float*)d_in[12];
  const float* p_w   = (const float*)d_in[13];
  const float* p_b   = (const float*)d_in[14];
  const float* p_g   = (const float*)d_in[15];
  const float* p_be  = (const float*)d_in[16];
  float* out = (float*)d_out;
  float* ws  = (float*)d_ws;

  k_init<<<1, 32, 0, stream>>>((int*)(ws + WS_CNT));
  k_nms<<<(NB * NHM + 255) / 256, 256, 0, stream>>>(hm, ws + WS_MASKED, ws + WS_FLAT,
                                                    (int*)(ws + WS_CNT));
  k_select<<<dim3(NB, 2), 256, 0, stream>>>(ws);
  k_combine<<<NB, 128, 0, stream>>>(ws, out);
  k_pool<<<dim3(NUMQ, NB), 256, 0, stream>>>(s0, s1, s2, s3, hm, lvlw, ws);
  k_content<<<50, 256, 0, stream>>>(ws, out, c_w1, c_b1, c_g1, c_be1, c_w2, c_b2);
  k_pos<<<50, 256, 0, stream>>>(ws, out, p_w, p_b, p_g, p_be);
}